// DFineInitialQueryAndReferenceGenerator_8839042695533
// MI455X (gfx1250) — compile-verified
//
#include <hip/hip_runtime.h>
#include <stdint.h>
#include <math.h>

#define BATCH 32
#define SEQ   8400
#define NLAB  80
#define DIM   256
#define NQ    300
#define NSORT 16384   // next pow2 >= SEQ

// Monotone map: float total order -> unsigned total order
__device__ __forceinline__ unsigned order_f32(float f) {
    unsigned u = __float_as_uint(f);
    return (u & 0x80000000u) ? ~u : (u | 0x80000000u);
}

// ---------------------------------------------------------------------------
// Kernel A: stream class logits through LDS with CDNA5 async global->LDS DMA,
// compute per-row max, emit 64-bit sortable keys (value-desc, index-asc ties).
// One thread <-> one row; one wave stages 32 contiguous rows (10240 B).
// ---------------------------------------------------------------------------
__global__ __launch_bounds__(256) void scores_kernel(
    const float* __restrict__ cls, unsigned long long* __restrict__ keys)
{
    // 8 waves/block * 32 rows * 80 floats = 81920 B of LDS
    __shared__ float tiles[256 * NLAB];

    const unsigned tid  = threadIdx.x;
    const unsigned lane = tid & 31u;
    const unsigned wave = tid >> 5;
    const unsigned g    = blockIdx.x * 256u + tid;   // global row id < 268800
    const unsigned wave_row0 = g & ~31u;             // first row of this wave's tile

    // LDS byte offset of the shared array: low 32 bits of its flat address
    // (LDS aperture keeps the workgroup-relative offset in addr[31:0]).
    const unsigned lds_array = (unsigned)(uintptr_t)(&tiles[0]);
    const unsigned lds_wave  = lds_array + wave * (32u * NLAB * 4u);   // +10240 per wave

    // Contiguous 10240-byte tile: rows [wave_row0, wave_row0+32) of the class tensor
    const unsigned long long gtile =
        (unsigned long long)(uintptr_t)cls + (unsigned long long)wave_row0 * (NLAB * 4u);

    // 20 async b128 copies: each instruction moves 32 lanes * 16 B = 512 B
    #pragma unroll
    for (int i = 0; i < 20; ++i) {
        unsigned           ldsaddr = lds_wave + (unsigned)i * 512u + lane * 16u;
        unsigned long long gaddr   = gtile + (unsigned long long)i * 512u + lane * 16u;
        asm volatile("global_load_async_to_lds_b128 %0, %1, off"
                     :
                     : "v"(ldsaddr), "v"(gaddr)
                     : "memory");
    }
    asm volatile("s_wait_asynccnt 0" ::: "memory");

    // Row-max from LDS (16B-aligned float4 reads; 4-way bank conflict, negligible)
    const float4* r4 =
        reinterpret_cast<const float4*>(&tiles[wave * 32u * NLAB + lane * NLAB]);
    float m = -3.402823466e38f;
    #pragma unroll
    for (int i = 0; i < NLAB / 4; ++i) {
        float4 v = r4[i];
        m = fmaxf(m, fmaxf(fmaxf(v.x, v.y), fmaxf(v.z, v.w)));
    }

    const unsigned s = g % SEQ;  // index within batch
    keys[g] = ((unsigned long long)order_f32(m) << 32) | (unsigned long long)(~s);
}

// ---------------------------------------------------------------------------
// Kernel B: per-batch exact top-300 via full bitonic sort (descending) of
// 16384 64-bit keys in 128 KB LDS. Emits top-300 original indices in order.
// ---------------------------------------------------------------------------
__global__ __launch_bounds__(512) void sort_kernel(
    const unsigned long long* __restrict__ keys, int* __restrict__ topk)
{
    __shared__ unsigned long long k[NSORT];   // 128 KB
    const unsigned b   = blockIdx.x;
    const unsigned tid = threadIdx.x;

    for (unsigned i = tid; i < NSORT; i += 512u)
        k[i] = (i < SEQ) ? keys[(size_t)b * SEQ + i] : 0ull;   // pad with -inf keys
    __syncthreads();

    for (unsigned kk = 2; kk <= NSORT; kk <<= 1) {
        for (unsigned j = kk >> 1; j > 0; j >>= 1) {
            for (unsigned i = tid; i < NSORT; i += 512u) {
                unsigned ixj = i ^ j;
                if (ixj > i) {
                    unsigned long long a = k[i], c = k[ixj];
                    bool up = ((i & kk) == 0);
                    // descending sort: "up" blocks keep larger element first
                    if (up ? (a < c) : (a > c)) { k[i] = c; k[ixj] = a; }
                }
            }
            __syncthreads();
        }
    }

    if (tid < NQ)
        topk[b * NQ + tid] = (int)(~(unsigned)(k[tid] & 0xFFFFFFFFull));
}

// ---------------------------------------------------------------------------
// Kernel C: gather coords (+sigmoid), memory rows, logits into concatenated
// output: [refs (B,NQ,4) | target (B,NQ,256) | logits (B,NQ,80) | boxes (B,NQ,4)]
// ---------------------------------------------------------------------------
__global__ __launch_bounds__(128) void gather_kernel(
    const float* __restrict__ cls, const float* __restrict__ coord,
    const float* __restrict__ mem, const int* __restrict__ topk,
    float* __restrict__ out)
{
    const unsigned bq  = blockIdx.x;          // 0 .. B*NQ-1
    const unsigned b   = bq / NQ;
    const unsigned q   = bq % NQ;
    const unsigned tid = threadIdx.x;
    const int idx = topk[b * NQ + q];

    float* o_ref = out;                                        // (B,NQ,4)
    float* o_tgt = out + (size_t)BATCH * NQ * 4;               // (B,NQ,DIM)
    float* o_log = o_tgt + (size_t)BATCH * NQ * DIM;           // (B,NQ,NLAB)
    float* o_box = o_log + (size_t)BATCH * NQ * NLAB;          // (B,NQ,4)

    const size_t src = (size_t)b * SEQ + (size_t)idx;
    const size_t dst = (size_t)b * NQ + q;

    if (tid < 4u) {
        float v = coord[src * 4 + tid];
        o_ref[dst * 4 + tid] = v;
        o_box[dst * 4 + tid] = 1.0f / (1.0f + expf(-v));
    }
    for (unsigned i = tid; i < DIM; i += 128u)
        o_tgt[dst * DIM + i] = mem[src * DIM + i];
    for (unsigned i = tid; i < NLAB; i += 128u)
        o_log[dst * NLAB + i] = cls[src * NLAB + i];
}

// ---------------------------------------------------------------------------
extern "C" void kernel_launch(void* const* d_in, const int* in_sizes, int n_in,
                              void* d_out, int out_size, void* d_ws, size_t ws_size,
                              hipStream_t stream) {
    const float* cls   = (const float*)d_in[0];   // (32, 8400, 80)
    const float* coord = (const float*)d_in[1];   // (32, 8400, 4)
    const float* mem   = (const float*)d_in[2];   // (32, 8400, 256)
    // d_in[3] (sources_last_element) is unused by the reference outputs.

    unsigned long long* keys = (unsigned long long*)d_ws;                 // 268800 * 8 B
    int* topk = (int*)((char*)d_ws + (size_t)BATCH * SEQ * sizeof(unsigned long long));
    float* out = (float*)d_out;

    scores_kernel<<<(BATCH * SEQ) / 256, 256, 0, stream>>>(cls, keys);
    sort_kernel<<<BATCH, 512, 0, stream>>>(keys, topk);
    gather_kernel<<<BATCH * NQ, 128, 0, stream>>>(cls, coord, mem, topk, out);
}